// SpatialQuadLayer_16432544875289
// MI455X (gfx1250) — compile-verified
//
#include <hip/hip_runtime.h>

typedef float v2f __attribute__((ext_vector_type(2)));
typedef float v8f __attribute__((ext_vector_type(8)));

#define QLEN 9
#define NF 54
#define NFP 56          // padded K (multiple of 4)
#define NCHUNK 14       // 56 / 4
#define CHN 32
#define BS 4
#define HH 128
#define WW 128
#define OC 4

// One wave = one image row (128 px) of one (b, c); 8 waves per block.
// Per 32-window segment: VALU builds 54 features/window, LDS re-layouts them
// into WMMA B fragments, 28x v_wmma_f32_16x16x4_f32 do the 4x54 projection.
__global__ __launch_bounds__(256)
void spatial_quad_kernel(const float* __restrict__ x,
                         const float* __restrict__ Wm,
                         const float* __restrict__ bias,
                         float* __restrict__ out)
{
    __shared__ float lds[8][32][NFP];          // 57,344 B

    const int lane = threadIdx.x & 31;
    const int wv   = threadIdx.x >> 5;
    const int row  = blockIdx.x * 8 + wv;      // 0 .. 16383
    const int b    = row >> 12;                // / (CHN*HH)
    const int c    = (row >> 7) & (CHN - 1);
    const int y    = row & (HH - 1);

    const float* __restrict__ xplane = x + ((size_t)(b * CHN + c) * HH) * WW;

    // ---- A fragment: W rows (M=0..3 live, 4..15 zero), K padded to 56 ----
    const int m     = lane & 15;               // M row (and B window-within-16)
    const int khalf = (lane < 16) ? 0 : 2;     // K sub-offset per ISA layout
    v2f afrag[NCHUNK];
    #pragma unroll
    for (int t = 0; t < NCHUNK; ++t) {
        const int k = 4 * t + khalf;
        afrag[t].x = (m < OC && k     < NF) ? Wm[m * NF + k]     : 0.0f;
        afrag[t].y = (m < OC && k + 1 < NF) ? Wm[m * NF + k + 1] : 0.0f;
    }

    for (int seg = 0; seg < 4; ++seg) {
        const int xp = seg * 32 + lane;        // this lane's window (pixel)

        // ---- gather 3x3 window with zero padding (selects, EXEC intact) ----
        float w[QLEN];
        #pragma unroll
        for (int ky = 0; ky < 3; ++ky) {
            const int yy = y + ky - 1;
            #pragma unroll
            for (int kx = 0; kx < 3; ++kx) {
                const int xx = xp + kx - 1;
                const bool ok = (yy >= 0) & (yy < HH) & (xx >= 0) & (xx < WW);
                w[ky * 3 + kx] = ok ? xplane[yy * WW + xx] : 0.0f;
            }
        }

        // ---- 54 features: 9 linear + 45 triu quadratic (np.triu order) ----
        float f[NFP];
        #pragma unroll
        for (int q = 0; q < QLEN; ++q) f[q] = w[q];
        {
            int t = QLEN;
            #pragma unroll
            for (int i = 0; i < QLEN; ++i)
                #pragma unroll
                for (int j = i; j < QLEN; ++j)
                    f[t++] = w[i] * w[j];
        }
        f[54] = 0.0f; f[55] = 0.0f;

        // ---- stage to LDS (b128 stores; 224 B window stride) ----
        #pragma unroll
        for (int g = 0; g < NCHUNK; ++g) {
            *(float4*)&lds[wv][lane][4 * g] =
                make_float4(f[4*g], f[4*g+1], f[4*g+2], f[4*g+3]);
        }
        __syncthreads();

        // ---- WMMA: D = A(W) x B(features) ; two 16-window tiles ----
        v8f acc0 = {};
        v8f acc1 = {};
        const float* __restrict__ baseA = &lds[wv][m][khalf];       // windows 0..15
        const float* __restrict__ baseB = &lds[wv][16 + m][khalf];  // windows 16..31
        #pragma unroll
        for (int g = 0; g < NCHUNK; ++g) {
            v2f bA = *(const v2f*)(baseA + 4 * g);
            v2f bB = *(const v2f*)(baseB + 4 * g);
            acc0 = __builtin_amdgcn_wmma_f32_16x16x4_f32(
                false, afrag[g], false, bA, (short)0, acc0, false, false);
            acc1 = __builtin_amdgcn_wmma_f32_16x16x4_f32(
                false, afrag[g], false, bB, (short)0, acc1, false, false);
        }
        __syncthreads();

        // ---- store: lanes 0..15 hold D rows 0..3 (out channels) ----
        if (lane < 16) {
            const int x0 = seg * 32 + lane;
            #pragma unroll
            for (int r = 0; r < OC; ++r) {
                const float bv = bias[r];
                const size_t o =
                    ((size_t)(b * (OC * CHN) + r * CHN + c) * HH + y) * WW;
                out[o + x0]      = acc0[r] + bv;
                out[o + x0 + 16] = acc1[r] + bv;
            }
        }
    }
}

extern "C" void kernel_launch(void* const* d_in, const int* in_sizes, int n_in,
                              void* d_out, int out_size, void* d_ws, size_t ws_size,
                              hipStream_t stream) {
    const float* x    = (const float*)d_in[0];
    const float* Wm   = (const float*)d_in[1];
    const float* bias = (const float*)d_in[2];
    float* out        = (float*)d_out;

    // rows total = 4 * 32 * 128 = 16384 ; 8 rows (waves) per block
    dim3 grid(16384 / 8);
    dim3 block(256);
    hipLaunchKernelGGL(spatial_quad_kernel, grid, block, 0, stream,
                       x, Wm, bias, out);
}